// Decoder_49280454754887
// MI455X (gfx1250) — compile-verified
//
#include <hip/hip_runtime.h>
#include <hip/hip_bf16.h>
#include <math.h>

// Problem constants (match reference)
#define BQ   16
#define TQ   64
#define HQ   1024
#define DQ   2048      // 2H
#define EMBQ 512
#define G3   3072      // 3H
#define CATQ 3584      // EMB + 3H
#define MXO  1024
#define VQ   32000

typedef float v2f __attribute__((ext_vector_type(2)));
typedef float v8f __attribute__((ext_vector_type(8)));

// ---------------------------------------------------------------------------
// Generic fp32 WMMA GEMM: C[M,N] = A[M,K] @ B[K,N] (+bias), all row-major.
// One wave (32 threads) computes (16*RT) rows x 64 cols using
// V_WMMA_F32_16X16X4_F32. Requires M % (16*RT) == 0, N % 64 == 0, K % 4 == 0.
// out_perm==1: rows are (t*B + b); store to Cout[(b*T + t)*ldc + col].
// ---------------------------------------------------------------------------
template <int RT>
__global__ __launch_bounds__(32) void gemm_wmma_f32(
    const float* __restrict__ A, int lda,
    const float* __restrict__ Bm, int ldb,
    const float* __restrict__ bias,
    float* __restrict__ Cout, int ldc,
    int K, int out_perm)
{
    const int lane = threadIdx.x & 31;
    const int hf   = lane >> 4;       // 0: lanes 0-15, 1: lanes 16-31
    const int l16  = lane & 15;
    const int col0 = blockIdx.x * 64; // 4 col-tiles of 16
    const int row0 = blockIdx.y * (16 * RT);

    v8f acc[RT][4];
#pragma unroll
    for (int r = 0; r < RT; ++r)
#pragma unroll
        for (int c = 0; c < 4; ++c) acc[r][c] = (v8f)0.0f;

    for (int k = 0; k < K; k += 4) {
        const int koff = k + hf * 2;  // lanes 16-31 hold K+2,K+3
        v2f a[RT];
#pragma unroll
        for (int r = 0; r < RT; ++r) {
            const float* ap = A + (size_t)(row0 + r * 16 + l16) * lda + koff;
            a[r].x = ap[0];
            a[r].y = ap[1];
        }
#pragma unroll
        for (int ct = 0; ct < 4; ++ct) {
            v2f b;
            const float* bp = Bm + (size_t)koff * ldb + (col0 + ct * 16 + l16);
            b.x = bp[0];
            b.y = bp[ldb];
#pragma unroll
            for (int r = 0; r < RT; ++r) {
                acc[r][ct] = __builtin_amdgcn_wmma_f32_16x16x4_f32(
                    false, a[r], false, b, (short)0, acc[r][ct], false, false);
            }
        }
    }

    // Epilogue: C/D layout -> VGPR i holds rows (i, i+8) split across halves.
#pragma unroll
    for (int r = 0; r < RT; ++r) {
#pragma unroll
        for (int ct = 0; ct < 4; ++ct) {
            const int col = col0 + ct * 16 + l16;
            const float bv = bias ? bias[col] : 0.0f;
#pragma unroll
            for (int i = 0; i < 8; ++i) {
                const int row = row0 + r * 16 + i + hf * 8;
                const float v = acc[r][ct][i] + bv;
                if (out_perm) {
                    const int t = row / BQ;
                    const int b = row % BQ;
                    Cout[(size_t)(b * TQ + t) * ldc + col] = v;
                } else {
                    Cout[(size_t)row * ldc + col] = v;
                }
            }
        }
    }
}

// ---------------------------------------------------------------------------
// Fused attention step: a = tanh(spre + b_s + he[:,t,:]); e = softmax(a);
// c[b,:] = sum_j e[b,j] * h[b,j,:]. One block per batch row b.
// ---------------------------------------------------------------------------
__global__ __launch_bounds__(256) void att_ctx_kernel(
    const float* __restrict__ he,    // [B*T, T]
    const float* __restrict__ spre,  // [B, T]
    const float* __restrict__ b_s,   // [T]
    const float* __restrict__ h,     // [B, T, D]
    float* __restrict__ Ct,          // [B, D] (C + t*B*D)
    int t)
{
    __shared__ float red[TQ];
    __shared__ float e[TQ];
    __shared__ float mval, sval;

    const int b   = blockIdx.x;
    const int tid = threadIdx.x;

    float a = 0.0f;
    if (tid < TQ) {
        a = tanhf(spre[b * TQ + tid] + b_s[tid] + he[(size_t)(b * TQ + t) * TQ + tid]);
        red[tid] = a;
    }
    __syncthreads();
    if (tid == 0) {
        float m = red[0];
        for (int j = 1; j < TQ; ++j) m = fmaxf(m, red[j]);
        mval = m;
    }
    __syncthreads();
    float ex = 0.0f;
    if (tid < TQ) {
        ex = __expf(a - mval);
        red[tid] = ex;
    }
    __syncthreads();
    if (tid == 0) {
        float s = 0.0f;
        for (int j = 0; j < TQ; ++j) s += red[j];
        sval = s;
    }
    __syncthreads();
    if (tid < TQ) e[tid] = ex / sval;
    __syncthreads();

    // context: 256 threads sweep D columns
    for (int d = tid; d < DQ; d += 256) {
        float acc = 0.0f;
#pragma unroll 8
        for (int j = 0; j < TQ; ++j)
            acc += e[j] * h[((size_t)b * TQ + j) * DQ + d];
        Ct[(size_t)b * DQ + d] = acc;
    }
}

// GRU elementwise (torch gate order r,z,n). Reads prev state, writes new
// state in place (per-thread element) and into the step archive S[t].
__global__ void gru_kernel(const float* __restrict__ gi,
                           const float* __restrict__ gh,
                           float* __restrict__ s_cur,  // [B,H], in/out
                           float* __restrict__ S_t)    // [B,H]
{
    const int idx = blockIdx.x * blockDim.x + threadIdx.x;
    if (idx >= BQ * HQ) return;
    const int b = idx / HQ, hh = idx % HQ;
    const size_t base = (size_t)b * G3;
    const float ir = gi[base + hh];
    const float iz = gi[base + HQ + hh];
    const float in_ = gi[base + 2 * HQ + hh];
    const float hr = gh[base + hh];
    const float hz = gh[base + HQ + hh];
    const float hn = gh[base + 2 * HQ + hh];
    const float r = 1.0f / (1.0f + __expf(-(ir + hr)));
    const float z = 1.0f / (1.0f + __expf(-(iz + hz)));
    const float n = tanhf(in_ + r * hn);
    const float sp = s_cur[idx];
    const float sn = (1.0f - z) * n + z * sp;
    s_cur[idx] = sn;
    S_t[idx]   = sn;
}

__global__ void zero_kernel(float* __restrict__ p, int n)
{
    const int idx = blockIdx.x * blockDim.x + threadIdx.x;
    if (idx < n) p[idx] = 0.0f;
}

// X[m] = concat(S[m] (H), Yemb[m] (EMB), C[m] (D)), rows m = t*B + b.
__global__ void concat_kernel(const float* __restrict__ S,
                              const float* __restrict__ Y,
                              const float* __restrict__ C,
                              float* __restrict__ X)
{
    const size_t total = (size_t)TQ * BQ * CATQ;
    for (size_t idx = (size_t)blockIdx.x * blockDim.x + threadIdx.x;
         idx < total; idx += (size_t)gridDim.x * blockDim.x) {
        const size_t row = idx / CATQ;
        const int c = (int)(idx % CATQ);
        float v;
        if (c < HQ)             v = S[row * HQ + c];
        else if (c < HQ + EMBQ) v = Y[row * EMBQ + (c - HQ)];
        else                    v = C[row * DQ + (c - HQ - EMBQ)];
        X[idx] = v;
    }
}

__global__ void maxout_kernel(const float* __restrict__ tt,  // [M, 2*MXO]
                              float* __restrict__ tm)        // [M, MXO]
{
    const size_t total = (size_t)TQ * BQ * MXO;
    for (size_t idx = (size_t)blockIdx.x * blockDim.x + threadIdx.x;
         idx < total; idx += (size_t)gridDim.x * blockDim.x) {
        const size_t row = idx / MXO;
        const int i = (int)(idx % MXO);
        tm[idx] = fmaxf(tt[row * (2 * MXO) + i], tt[row * (2 * MXO) + MXO + i]);
    }
}

// ---------------------------------------------------------------------------
extern "C" void kernel_launch(void* const* d_in, const int* in_sizes, int n_in,
                              void* d_out, int out_size, void* d_ws, size_t ws_size,
                              hipStream_t stream)
{
    const float* h     = (const float*)d_in[0];   // [B,T,D]
    const float* W_h   = (const float*)d_in[1];   // [D,T]
    const float* b_h   = (const float*)d_in[2];   // [T]
    const float* W_s   = (const float*)d_in[3];   // [H,T]
    const float* b_s   = (const float*)d_in[4];   // [T]
    const float* W_ih  = (const float*)d_in[5];   // [D,3H]
    const float* b_ih  = (const float*)d_in[6];   // [3H]
    const float* W_hh  = (const float*)d_in[7];   // [H,3H]
    const float* b_hh  = (const float*)d_in[8];   // [3H]
    const float* W_emb = (const float*)d_in[9];   // [H,EMB]
    const float* b_emb = (const float*)d_in[10];  // [EMB]
    const float* W_t   = (const float*)d_in[11];  // [CAT,2*MXO]
    const float* b_t   = (const float*)d_in[12];  // [2*MXO]
    const float* W_out = (const float*)d_in[13];  // [MXO,V]
    const float* b_out = (const float*)d_in[14];  // [V]
    float* out = (float*)d_out;                   // [B,T,V]

    // Workspace layout (floats)
    float* ws = (float*)d_ws;
    size_t off = 0;
    float* he    = ws + off; off += (size_t)BQ * TQ * TQ;     // [B*T, T]
    float* S     = ws + off; off += (size_t)TQ * BQ * HQ;     // [T*B, H]
    float* C     = ws + off; off += (size_t)TQ * BQ * DQ;     // [T*B, D]
    float* s_cur = ws + off; off += (size_t)BQ * HQ;          // [B, H]
    float* spre  = ws + off; off += (size_t)BQ * TQ;          // [B, T]
    float* gi    = ws + off; off += (size_t)BQ * G3;          // [B, 3H]
    float* gh    = ws + off; off += (size_t)BQ * G3;          // [B, 3H]
    float* yemb  = ws + off; off += (size_t)TQ * BQ * EMBQ;   // [T*B, EMB]
    float* X     = ws + off; off += (size_t)TQ * BQ * CATQ;   // [T*B, CAT]
    float* tt    = ws + off; off += (size_t)TQ * BQ * 2 * MXO;// [T*B, 2*MXO]
    float* tmax  = ws + off; off += (size_t)TQ * BQ * MXO;    // [T*B, MXO]
    (void)ws_size; (void)in_sizes; (void)n_in; (void)out_size;

    // s0 = 0
    zero_kernel<<<(BQ * HQ + 255) / 256, 256, 0, stream>>>(s_cur, BQ * HQ);

    // he = h_flat @ W_h + b_h : [1024,2048]@[2048,64]
    gemm_wmma_f32<4><<<dim3(TQ / 64, (BQ * TQ) / 64), 32, 0, stream>>>(
        h, DQ, W_h, TQ, b_h, he, TQ, DQ, 0);

    // Serial recurrence over T steps
    for (int t = 0; t < TQ; ++t) {
        // spre = s @ W_s : [16,1024]@[1024,64] (b_s added inside att kernel)
        gemm_wmma_f32<1><<<dim3(TQ / 64, 1), 32, 0, stream>>>(
            s_cur, HQ, W_s, TQ, nullptr, spre, TQ, HQ, 0);

        // attention + softmax + context -> C[t]
        att_ctx_kernel<<<BQ, 256, 0, stream>>>(he, spre, b_s, h,
                                               C + (size_t)t * BQ * DQ, t);

        // gi = c @ W_ih + b_ih : [16,2048]@[2048,3072]
        gemm_wmma_f32<1><<<dim3(G3 / 64, 1), 32, 0, stream>>>(
            C + (size_t)t * BQ * DQ, DQ, W_ih, G3, b_ih, gi, G3, DQ, 0);

        // gh = s @ W_hh + b_hh : [16,1024]@[1024,3072]
        gemm_wmma_f32<1><<<dim3(G3 / 64, 1), 32, 0, stream>>>(
            s_cur, HQ, W_hh, G3, b_hh, gh, G3, HQ, 0);

        // GRU update -> s_cur, S[t]
        gru_kernel<<<(BQ * HQ + 255) / 256, 256, 0, stream>>>(
            gi, gh, s_cur, S + (size_t)t * BQ * HQ);
    }

    // ------- Batched tail over M = T*B = 1024 rows (hoisted out of scan) ----
    // yemb = S @ W_emb + b_emb : [1024,1024]@[1024,512]
    gemm_wmma_f32<4><<<dim3(EMBQ / 64, (TQ * BQ) / 64), 32, 0, stream>>>(
        S, HQ, W_emb, EMBQ, b_emb, yemb, EMBQ, HQ, 0);

    // X = [S | yemb | C]
    concat_kernel<<<4096, 256, 0, stream>>>(S, yemb, C, X);

    // tt = X @ W_t + b_t : [1024,3584]@[3584,2048]
    gemm_wmma_f32<4><<<dim3((2 * MXO) / 64, (TQ * BQ) / 64), 32, 0, stream>>>(
        X, CATQ, W_t, 2 * MXO, b_t, tt, 2 * MXO, CATQ, 0);

    // maxout
    maxout_kernel<<<2048, 256, 0, stream>>>(tt, tmax);

    // out = tmax @ W_out + b_out : [1024,1024]@[1024,32000], permuted store
    gemm_wmma_f32<4><<<dim3(VQ / 64, (TQ * BQ) / 64), 32, 0, stream>>>(
        tmax, MXO, W_out, VQ, b_out, out, VQ, MXO, 1);
}